// PointNet_53188874994270
// MI455X (gfx1250) — compile-verified
//
#include <hip/hip_runtime.h>
#include <hip/hip_bf16.h>

typedef __attribute__((ext_vector_type(2))) float v2f;
typedef __attribute__((ext_vector_type(8))) float v8f;

#define BN_EPS 1e-5f

static __device__ __forceinline__ unsigned ballot32(bool p) {
#if __has_builtin(__builtin_amdgcn_ballot_w32)
  return __builtin_amdgcn_ballot_w32(p);
#else
  return (unsigned)__ballot(p);
#endif
}

// ---------------------------------------------------------------------------
// Farthest point sampling: one workgroup per batch. dists live in registers
// (P per thread), centroid broadcast through LDS, argmax via LDS tree.
// Also emits new_xyz directly (the selected centroid coords).
// ---------------------------------------------------------------------------
template <int THREADS, int P>
__global__ __launch_bounds__(THREADS) void fps_kernel(
    const float* __restrict__ xyz, int N, int npoint,
    int* __restrict__ out_idx, float* __restrict__ out_xyz) {
  __shared__ float s_best[THREADS];
  __shared__ int s_bidx[THREADS];
  __shared__ float s_cent[3];
  const int b = blockIdx.x;
  const int t = threadIdx.x;
  const float* xb = xyz + (size_t)b * N * 3;

  float dist[P];
#pragma unroll
  for (int j = 0; j < P; ++j) dist[j] = 1e10f;

  int farthest = 0;
  for (int it = 0; it < npoint; ++it) {
    if (t == 0) {
      const float cx = xb[farthest * 3 + 0];
      const float cy = xb[farthest * 3 + 1];
      const float cz = xb[farthest * 3 + 2];
      s_cent[0] = cx; s_cent[1] = cy; s_cent[2] = cz;
      out_idx[b * npoint + it] = farthest;
      out_xyz[((size_t)b * npoint + it) * 3 + 0] = cx;
      out_xyz[((size_t)b * npoint + it) * 3 + 1] = cy;
      out_xyz[((size_t)b * npoint + it) * 3 + 2] = cz;
    }
    __syncthreads();
    const float cx = s_cent[0], cy = s_cent[1], cz = s_cent[2];
    float best = -1.0f;
    int bidx = t;
#pragma unroll
    for (int j = 0; j < P; ++j) {
      const int p = t + j * THREADS;
      const float dx = xb[p * 3 + 0] - cx;
      const float dy = xb[p * 3 + 1] - cy;
      const float dz = xb[p * 3 + 2] - cz;
      const float d = dx * dx + dy * dy + dz * dz;
      const float nd = fminf(dist[j], d);
      dist[j] = nd;
      if (nd > best) { best = nd; bidx = p; }
    }
    s_best[t] = best;
    s_bidx[t] = bidx;
    __syncthreads();
    for (int s = THREADS >> 1; s > 0; s >>= 1) {
      if (t < s) {
        if (s_best[t + s] > s_best[t]) {
          s_best[t] = s_best[t + s];
          s_bidx[t] = s_bidx[t + s];
        }
      }
      __syncthreads();
    }
    farthest = s_bidx[0];
    __syncthreads();
  }
}

// ---------------------------------------------------------------------------
// Ball query: one wave32 per centroid. Scan points in index order, ballot the
// in-radius lanes, peel set bits with ctz -> exactly the reference's
// "first nsample in-radius indices ascending, fill with first" semantics.
// ---------------------------------------------------------------------------
__global__ void ball_query_kernel(const float* __restrict__ xyz,
                                  const float* __restrict__ new_xyz, float r2,
                                  int B, int N, int S,
                                  int* __restrict__ out_idx) {
  const int wid = (int)((blockIdx.x * blockDim.x + threadIdx.x) >> 5);
  const int lane = threadIdx.x & 31;
  if (wid >= B * S) return;
  const int b = wid / S;
  const int s = wid % S;
  const float* xb = xyz + (size_t)b * N * 3;
  const float cx = new_xyz[((size_t)b * S + s) * 3 + 0];
  const float cy = new_xyz[((size_t)b * S + s) * 3 + 1];
  const float cz = new_xyz[((size_t)b * S + s) * 3 + 2];

  int idx4[4] = {N - 1, N - 1, N - 1, N - 1};
  int cnt = 0;
  for (int base = 0; base < N && cnt < 4; base += 32) {
    const int p = base + lane;
    bool in = false;
    if (p < N) {
      const float dx = xb[p * 3 + 0] - cx;
      const float dy = xb[p * 3 + 1] - cy;
      const float dz = xb[p * 3 + 2] - cz;
      in = (dx * dx + dy * dy + dz * dz) < r2;
    }
    unsigned m = ballot32(in);
    while (m && cnt < 4) {
      const int bp = __builtin_ctz(m);
      idx4[cnt++] = base + bp;
      m &= m - 1;
    }
  }
  if (cnt > 0) {
    for (int k = cnt; k < 4; ++k) idx4[k] = idx4[0];
  }
  if (lane == 0) {
    for (int k = 0; k < 4; ++k)
      out_idx[((size_t)b * S + s) * 4 + k] = idx4[k];
  }
}

// ---------------------------------------------------------------------------
// Build X[c][m] (m = (b*S+s)*K + k): rows 0..2 = gathered xyz - center,
// rows 3..3+C-1 = gathered feats, rows up to Cin4 zero-padded for WMMA K=4.
// ---------------------------------------------------------------------------
__global__ void pack_group_kernel(const float* __restrict__ xyz,
                                  const float* __restrict__ feats,
                                  const float* __restrict__ new_xyz,
                                  const int* __restrict__ idx,
                                  float* __restrict__ X, int B, int N, int S,
                                  int K, int C, int Cin4) {
  const int M = B * S * K;
  const int total = Cin4 * M;
  for (int i = blockIdx.x * blockDim.x + threadIdx.x; i < total;
       i += gridDim.x * blockDim.x) {
    const int c = i / M;
    const int m = i % M;
    const int k = m % K;
    const int s = (m / K) % S;
    const int b = m / (K * S);
    float v = 0.0f;
    const int pi = idx[((size_t)b * S + s) * 4 + k];
    if (c < 3) {
      v = xyz[((size_t)b * N + pi) * 3 + c] -
          new_xyz[((size_t)b * S + s) * 3 + c];
    } else if (c < 3 + C) {
      v = feats[((size_t)b * C + (c - 3)) * N + pi];
    }
    X[i] = v;
  }
}

// Stage 3 pack: X[c][m], m = b*Sn + n; rows 0..2 = l2_xyz, 3..3+C-1 = l2_feat.
__global__ void pack_stage3_kernel(const float* __restrict__ l2xyz,
                                   const float* __restrict__ l2feat,
                                   float* __restrict__ X, int B, int Sn, int C,
                                   int Cin4) {
  const int M = B * Sn;
  const int total = Cin4 * M;
  for (int i = blockIdx.x * blockDim.x + threadIdx.x; i < total;
       i += gridDim.x * blockDim.x) {
    const int c = i / M;
    const int m = i % M;
    const int n = m % Sn;
    const int b = m / Sn;
    float v = 0.0f;
    if (c < 3) v = l2xyz[((size_t)b * Sn + n) * 3 + c];
    else if (c < 3 + C) v = l2feat[((size_t)b * C + (c - 3)) * Sn + n];
    X[i] = v;
  }
}

// ---------------------------------------------------------------------------
// GEMM  Y[o][m] = sum_c W[o][c] * X[c][m] + bias[o]  via V_WMMA_F32_16X16X4_F32.
// One 16x16 D tile per wave. A = W tile (16x4), B = X tile (4x16).
// ISA layout: A lane = row (lane&15); VGPR0 holds K = 2*half, VGPR1 K = 2*half+1.
// D: VGPR j -> row o0 + j + 8*half, col m0 + (lane&15).
// Cout, M are multiples of 16; X is zero-padded to Cin4 (mult of 4);
// W loads are predicated on c < Cin.
// ---------------------------------------------------------------------------
__global__ __launch_bounds__(256) void gemm_bias_wmma(
    const float* __restrict__ W, const float* __restrict__ bias,
    const float* __restrict__ X, float* __restrict__ Y, int Cout, int Cin,
    int Cin4, int M) {
  const int lane = threadIdx.x & 31;
  const int half = lane >> 4;
  const int l15 = lane & 15;
  const int tilesM = M >> 4;
  const int tiles = (Cout >> 4) * tilesM;
  const int wid = (int)((blockIdx.x * blockDim.x + threadIdx.x) >> 5);
  const int nw = (int)((gridDim.x * blockDim.x) >> 5);
  for (int tile = wid; tile < tiles; tile += nw) {
    const int o0 = (tile / tilesM) << 4;
    const int m0 = (tile % tilesM) << 4;
    v8f acc = {};
#if __has_builtin(__builtin_amdgcn_wmma_f32_16x16x4_f32)
    const int arow = o0 + l15;
    for (int c = 0; c < Cin4; c += 4) {
      const int ka = c + (half << 1);
      v2f a, bm;
      a.x = (ka < Cin) ? W[(size_t)arow * Cin + ka] : 0.0f;
      a.y = (ka + 1 < Cin) ? W[(size_t)arow * Cin + ka + 1] : 0.0f;
      bm.x = X[(size_t)ka * M + m0 + l15];
      bm.y = X[(size_t)(ka + 1) * M + m0 + l15];
      if (c + 4 < Cin4)
        __builtin_prefetch(&X[(size_t)(ka + 4) * M + m0 + l15], 0, 1);
      acc = __builtin_amdgcn_wmma_f32_16x16x4_f32(
          false, a, false, bm, (short)0, acc, false, false);
    }
#else
    for (int c = 0; c < Cin; ++c) {
      const float xv = X[(size_t)c * M + m0 + l15];
#pragma unroll
      for (int j = 0; j < 8; ++j)
        acc[j] += W[(size_t)(o0 + j + (half ? 8 : 0)) * Cin + c] * xv;
    }
#endif
#pragma unroll
    for (int j = 0; j < 8; ++j) {
      const int row = o0 + j + (half ? 8 : 0);
      Y[(size_t)row * M + m0 + l15] = acc[j] + bias[row];
    }
  }
}

// ---------------------------------------------------------------------------
// BatchNorm (stats over all m, per channel) + ReLU, in place. Block = channel.
// ---------------------------------------------------------------------------
__global__ __launch_bounds__(256) void bn_relu_kernel(
    float* __restrict__ Y, const float* __restrict__ g,
    const float* __restrict__ be, int M) {
  __shared__ float s1[256];
  __shared__ float s2[256];
  const int o = blockIdx.x;
  const int t = threadIdx.x;
  float a = 0.0f, q = 0.0f;
  for (int m = t; m < M; m += 256) {
    const float v = Y[(size_t)o * M + m];
    a += v;
    q += v * v;
  }
  s1[t] = a;
  s2[t] = q;
  __syncthreads();
  for (int s = 128; s > 0; s >>= 1) {
    if (t < s) { s1[t] += s1[t + s]; s2[t] += s2[t + s]; }
    __syncthreads();
  }
  const float mean = s1[0] / (float)M;
  const float var = s2[0] / (float)M - mean * mean;
  const float inv = rsqrtf(var + BN_EPS);
  const float sc = g[o] * inv;
  const float sh = be[o] - mean * sc;
  for (int m = t; m < M; m += 256) {
    const float v = Y[(size_t)o * M + m] * sc + sh;
    Y[(size_t)o * M + m] = fmaxf(v, 0.0f);
  }
}

// Max over k (nsample) -> feat[b][o][s]
__global__ void maxpool_k_kernel(const float* __restrict__ Y,
                                 float* __restrict__ out, int B, int S, int K,
                                 int Cout) {
  const int M = B * S * K;
  const int total = B * Cout * S;
  for (int i = blockIdx.x * blockDim.x + threadIdx.x; i < total;
       i += gridDim.x * blockDim.x) {
    const int s = i % S;
    const int o = (i / S) % Cout;
    const int b = i / (S * Cout);
    const float* p = Y + (size_t)o * M + ((size_t)b * S + s) * K;
    float v = p[0];
    for (int k = 1; k < K; ++k) v = fmaxf(v, p[k]);
    out[((size_t)b * Cout + o) * S + s] = v;
  }
}

// Final max over the Sn points of each batch -> out[b][o]
__global__ void final_max_kernel(const float* __restrict__ Y,
                                 float* __restrict__ out, int B, int Sn,
                                 int Cout) {
  const int total = B * Cout;
  for (int i = blockIdx.x * blockDim.x + threadIdx.x; i < total;
       i += gridDim.x * blockDim.x) {
    const int o = i % Cout;
    const int b = i / Cout;
    const float* p = Y + (size_t)o * (B * Sn) + (size_t)b * Sn;
    float v = p[0];
    for (int n = 1; n < Sn; ++n) v = fmaxf(v, p[n]);
    out[(size_t)b * Cout + o] = v;
  }
}

static inline int cdiv(int a, int b) { return (a + b - 1) / b; }

extern "C" void kernel_launch(void* const* d_in, const int* in_sizes, int n_in,
                              void* d_out, int out_size, void* d_ws,
                              size_t ws_size, hipStream_t stream) {
  (void)in_sizes; (void)n_in; (void)out_size; (void)ws_size;
  const float* xyz = (const float*)d_in[0];   // (8, 32768, 3)
  const float* feat = (const float*)d_in[1];  // (8, 3, 32768)
  // p1/p2/p3 flattened: each layer = (W, b, g, be)
  const float* W1a = (const float*)d_in[2];
  const float* b1a = (const float*)d_in[3];
  const float* g1a = (const float*)d_in[4];
  const float* e1a = (const float*)d_in[5];
  const float* W1b = (const float*)d_in[6];
  const float* b1b = (const float*)d_in[7];
  const float* g1b = (const float*)d_in[8];
  const float* e1b = (const float*)d_in[9];
  const float* W2a = (const float*)d_in[10];
  const float* b2a = (const float*)d_in[11];
  const float* g2a = (const float*)d_in[12];
  const float* e2a = (const float*)d_in[13];
  const float* W2b = (const float*)d_in[14];
  const float* b2b = (const float*)d_in[15];
  const float* g2b = (const float*)d_in[16];
  const float* e2b = (const float*)d_in[17];
  const float* W3a = (const float*)d_in[18];
  const float* b3a = (const float*)d_in[19];
  const float* g3a = (const float*)d_in[20];
  const float* e3a = (const float*)d_in[21];
  const float* W3b = (const float*)d_in[22];
  const float* b3b = (const float*)d_in[23];
  const float* g3b = (const float*)d_in[24];
  const float* e3b = (const float*)d_in[25];
  float* out = (float*)d_out;

  const int B = 8, N = 32768;
  const int S1 = 128, S2 = 32, K = 4, Sn = 32;
  const int M1 = B * S1 * K;  // 4096
  const int M2 = B * S2 * K;  // 1024
  const int M3 = B * Sn;      // 256

  char* ws = (char*)d_ws;
  size_t off = 0;
  auto alloc = [&](size_t bytes) -> void* {
    void* p = (void*)(ws + off);
    off += (bytes + 255) & ~(size_t)255;
    return p;
  };
  int* fpsidx1 = (int*)alloc((size_t)B * S1 * 4);
  float* nxyz1 = (float*)alloc((size_t)B * S1 * 3 * 4);
  int* idx1 = (int*)alloc((size_t)B * S1 * 4 * 4);
  float* X1 = (float*)alloc((size_t)8 * M1 * 4);
  float* Y1a = (float*)alloc((size_t)64 * M1 * 4);
  float* Y1b = (float*)alloc((size_t)128 * M1 * 4);
  float* l1feat = (float*)alloc((size_t)B * 128 * S1 * 4);
  int* fpsidx2 = (int*)alloc((size_t)B * S2 * 4);
  float* nxyz2 = (float*)alloc((size_t)B * S2 * 3 * 4);
  int* idx2 = (int*)alloc((size_t)B * S2 * 4 * 4);
  float* X2 = (float*)alloc((size_t)132 * M2 * 4);
  float* Y2a = (float*)alloc((size_t)128 * M2 * 4);
  float* Y2b = (float*)alloc((size_t)256 * M2 * 4);
  float* l2feat = (float*)alloc((size_t)B * 256 * S2 * 4);
  float* X3 = (float*)alloc((size_t)260 * M3 * 4);
  float* Y3a = (float*)alloc((size_t)256 * M3 * 4);
  float* Y3b = (float*)alloc((size_t)512 * M3 * 4);

  // ---------------- Stage 1 ----------------
  fps_kernel<1024, 32><<<B, 1024, 0, stream>>>(xyz, N, S1, fpsidx1, nxyz1);
  ball_query_kernel<<<cdiv(B * S1, 8), 256, 0, stream>>>(
      xyz, nxyz1, 0.25f * 0.25f, B, N, S1, idx1);
  pack_group_kernel<<<cdiv(8 * M1, 256), 256, 0, stream>>>(
      xyz, feat, nxyz1, idx1, X1, B, N, S1, K, 3, 8);
  gemm_bias_wmma<<<cdiv((64 / 16) * (M1 / 16), 8), 256, 0, stream>>>(
      W1a, b1a, X1, Y1a, 64, 6, 8, M1);
  bn_relu_kernel<<<64, 256, 0, stream>>>(Y1a, g1a, e1a, M1);
  gemm_bias_wmma<<<cdiv((128 / 16) * (M1 / 16), 8), 256, 0, stream>>>(
      W1b, b1b, Y1a, Y1b, 128, 64, 64, M1);
  bn_relu_kernel<<<128, 256, 0, stream>>>(Y1b, g1b, e1b, M1);
  maxpool_k_kernel<<<cdiv(B * 128 * S1, 256), 256, 0, stream>>>(
      Y1b, l1feat, B, S1, K, 128);

  // ---------------- Stage 2 ----------------
  fps_kernel<128, 1><<<B, 128, 0, stream>>>(nxyz1, S1, S2, fpsidx2, nxyz2);
  ball_query_kernel<<<cdiv(B * S2, 8), 256, 0, stream>>>(
      nxyz1, nxyz2, 0.4f * 0.4f, B, S1, S2, idx2);
  pack_group_kernel<<<cdiv(132 * M2, 256), 256, 0, stream>>>(
      nxyz1, l1feat, nxyz2, idx2, X2, B, S1, S2, K, 128, 132);
  gemm_bias_wmma<<<cdiv((128 / 16) * (M2 / 16), 8), 256, 0, stream>>>(
      W2a, b2a, X2, Y2a, 128, 131, 132, M2);
  bn_relu_kernel<<<128, 256, 0, stream>>>(Y2a, g2a, e2a, M2);
  gemm_bias_wmma<<<cdiv((256 / 16) * (M2 / 16), 8), 256, 0, stream>>>(
      W2b, b2b, Y2a, Y2b, 256, 128, 128, M2);
  bn_relu_kernel<<<256, 256, 0, stream>>>(Y2b, g2b, e2b, M2);
  maxpool_k_kernel<<<cdiv(B * 256 * S2, 256), 256, 0, stream>>>(
      Y2b, l2feat, B, S2, K, 256);

  // ---------------- Stage 3 (group-all) ----------------
  pack_stage3_kernel<<<cdiv(260 * M3, 256), 256, 0, stream>>>(
      nxyz2, l2feat, X3, B, Sn, 256, 260);
  gemm_bias_wmma<<<cdiv((256 / 16) * (M3 / 16), 8), 256, 0, stream>>>(
      W3a, b3a, X3, Y3a, 256, 259, 260, M3);
  bn_relu_kernel<<<256, 256, 0, stream>>>(Y3a, g3a, e3a, M3);
  gemm_bias_wmma<<<cdiv((512 / 16) * (M3 / 16), 8), 256, 0, stream>>>(
      W3b, b3b, Y3a, Y3b, 512, 256, 256, M3);
  bn_relu_kernel<<<512, 256, 0, stream>>>(Y3b, g3b, e3b, M3);
  final_max_kernel<<<cdiv(B * 512, 256), 256, 0, stream>>>(
      Y3b, out, B, Sn, 512);
}